// CustomModel_88038239634077
// MI455X (gfx1250) — compile-verified
//
#include <hip/hip_runtime.h>

typedef float f4 __attribute__((ext_vector_type(4)));
typedef unsigned int v4u __attribute__((ext_vector_type(4)));
typedef int v4i __attribute__((ext_vector_type(4)));
typedef int v8i __attribute__((ext_vector_type(8)));

#define BUCKETS 50
#define DIM 128
#define TBL_ELEMS (BUCKETS * DIM)        // 6400 floats per table
#define TBL_TOTAL (2 * TBL_ELEMS)        // 12800 floats = 51200 bytes (S then T)
#define WAVES 8
#define BLOCK 256

// ---------------- Kernel 1: fold m1/v1/m2/v2 into scale/shift tables ----------------
__global__ void fds_precompute(const float* __restrict__ m1, const float* __restrict__ v1,
                               const float* __restrict__ m2, const float* __restrict__ v2,
                               float* __restrict__ ws) {
  int i = blockIdx.x * blockDim.x + threadIdx.x;
  if (i >= TBL_ELEMS) return;
  float m1v = m1[i], v1v = v1[i], m2v = m2[i], v2v = v2[i];
  float v1p = fmaxf(v1v, 0.0f);
  float v2p = fmaxf(v2v, 0.0f);
  float factor = v2p / fmaxf(v1p, 1e-12f);
  factor = fminf(fmaxf(factor, 0.1f), 10.0f);
  float s = sqrtf(factor);
  float S, T;
  if (v1p > 0.0f) { S = s; T = fmaf(-m1v, s, m2v); }  // (f-m1)*s+m2 == f*s + (m2-m1*s)
  else            { S = 1.0f; T = 0.0f; }             // v1==0 -> passthrough
  ws[i] = S;
  ws[TBL_ELEMS + i] = T;
}

// ---------------- Kernel 2: persistent streaming apply ----------------
__global__ void __launch_bounds__(BLOCK) fds_apply(const float* __restrict__ features,
                                                   const float* __restrict__ labels,
                                                   const float* __restrict__ tbl,
                                                   float* __restrict__ out,
                                                   long long N) {
  __shared__ __align__(16) float lds[TBL_TOTAL];

  const int wave = threadIdx.x >> 5;   // wave32
  const int lane = threadIdx.x & 31;

#if __has_builtin(__builtin_amdgcn_tensor_load_to_lds) && __has_builtin(__builtin_amdgcn_s_wait_tensorcnt)
  if (wave == 0) {
    // Tensor Data Mover: one 1-D DMA of the whole 51.2KB S|T blob into LDS.
    unsigned long long ga = (unsigned long long)tbl;
    unsigned lds_off = (unsigned)(unsigned long long)(void*)lds; // low 32 bits of flat addr == LDS byte offset
    v4u g0;
    g0.x = 1u;                                                   // count=1 (valid user descriptor)
    g0.y = lds_off;                                              // lds_addr
    g0.z = (unsigned)ga;                                         // global_addr[31:0]
    g0.w = (unsigned)((ga >> 32) & 0x01FFFFFFu) | 0x80000000u;   // global_addr[56:32] | type=2 (bits 127:126)
    v8i g1;
    g1[0] = 0x00020000;                                 // workgroup_mask=0, data_size=2 (4 bytes)
    g1[1] = (int)((TBL_TOTAL & 0xFFFF) << 16);          // tensor_dim0[15:0] @ bits 63:48
    g1[2] = (int)(((unsigned)TBL_TOTAL >> 16) | (1u << 16)); // tensor_dim0[31:16] | tensor_dim1=1
    g1[3] = (int)((TBL_TOTAL & 0xFFFF) << 16);          // tile_dim0 @ bits 127:112
    g1[4] = 0;                                          // tile_dim1=0, tile_dim2=0 (unused)
    g1[5] = TBL_TOTAL;                                  // tensor_dim0_stride[31:0]
    g1[6] = 0;
    g1[7] = 0;
    v4i z4 = {0, 0, 0, 0};
    v8i z8 = {0, 0, 0, 0, 0, 0, 0, 0};
    // 6-arg form on this toolchain: (g0 u32x4, g1 i32x8, g2 i32x4, g3 i32x4, i32x8, cpol)
    __builtin_amdgcn_tensor_load_to_lds(g0, g1, z4, z4, z8, 0);
    __builtin_amdgcn_s_wait_tensorcnt(0);
  }
#else
  // Fallback: cooperative vectorized copy
  {
    const f4* src4 = (const f4*)tbl;
    f4* dst4 = (f4*)lds;
    for (int i = threadIdx.x; i < TBL_TOTAL / 4; i += BLOCK) dst4[i] = src4[i];
  }
#endif
  __syncthreads();

  const f4* __restrict__ feat4 = (const f4*)features;
  f4* __restrict__ out4 = (f4*)out;
  const f4* __restrict__ S4 = (const f4*)lds;
  const f4* __restrict__ T4 = (const f4*)(lds + TBL_ELEMS);

  // One wave per row: 32 lanes x float4 = 128 columns.
  const long long stride = (long long)gridDim.x * WAVES;
  for (long long row = (long long)blockIdx.x * WAVES + wave; row < N; row += stride) {
    float lab = labels[row];
    lab = fminf(fmaxf(lab, 0.0f), 1.0f);
    int b = (int)(lab * (float)BUCKETS);      // truncation matches astype(int32) for lab>=0
    b = b > (BUCKETS - 1) ? (BUCKETS - 1) : b;

    const long long base = row * (DIM / 4) + lane;
    f4 f = __builtin_nontemporal_load(&feat4[base]);  // streamed once: keep out of caches
    f4 s = S4[b * (DIM / 4) + lane];                  // ds_load_b128, conflict-free
    f4 t = T4[b * (DIM / 4) + lane];
    f4 r = f * s + t;                                 // v_fma per component
    __builtin_nontemporal_store(r, &out4[base]);
  }
}

extern "C" void kernel_launch(void* const* d_in, const int* in_sizes, int n_in,
                              void* d_out, int out_size, void* d_ws, size_t ws_size,
                              hipStream_t stream) {
  const float* features = (const float*)d_in[0];
  const float* labels   = (const float*)d_in[1];
  const float* m1 = (const float*)d_in[2];
  const float* v1 = (const float*)d_in[3];
  const float* m2 = (const float*)d_in[4];
  const float* v2 = (const float*)d_in[5];
  float* out = (float*)d_out;
  float* ws  = (float*)d_ws;   // 51200 bytes used: S table then T table

  long long N = in_sizes[1];   // labels count = number of rows

  fds_precompute<<<(TBL_ELEMS + BLOCK - 1) / BLOCK, BLOCK, 0, stream>>>(m1, v1, m2, v2, ws);

  long long maxBlocks = (N + WAVES - 1) / WAVES;
  int grid = 2048;                       // persistent blocks: amortize LDS staging
  if (maxBlocks < (long long)grid) grid = (int)maxBlocks;
  if (grid < 1) grid = 1;
  fds_apply<<<grid, BLOCK, 0, stream>>>(features, labels, ws, out, N);
}